// QuestionFlowLayer_910533067614
// MI455X (gfx1250) — compile-verified
//
#include <hip/hip_runtime.h>

// ---------------------------------------------------------------------------
// QuestionFlowLayer: out = concat([q, exclusive_prefix_mean(q)], last axis)
// q: [N=1024, L=64, D=1024] f32  ->  out: [1024, 64, 2048] f32
//
// Memory-bound (768 MB @ 23.3 TB/s ~ 33us). Intra-chunk prefix sums via
// V_WMMA_F32_16X16X4_F32 (strictly-lower-triangular A x data B), carry
// propagated per 16-turn chunk via a wave32 shuffle. NT cache hints for the
// streamed data; prefetch for the serial chunk chain. Division replaced by
// v_rcp_f32 + v_max (no divergence: exclusive prefix at t==0 is exactly 0).
// ---------------------------------------------------------------------------

typedef float v2f __attribute__((ext_vector_type(2)));
typedef float v8f __attribute__((ext_vector_type(8)));

namespace {
constexpr int kN      = 1024;          // turns
constexpr int kL      = 64;
constexpr int kD      = 1024;
constexpr int kC      = kL * kD;       // 65536 flattened columns (l*1024 + d)
constexpr int kOutRow = kL * 2 * kD;   // 131072 f32 per output turn
constexpr int kChunks = kN / 16;       // 64 chunks of 16 turns
constexpr int kWaves  = kC / 16;       // 4096 waves, 16 columns each
}

__global__ __launch_bounds__(256)
void qflow_wmma_scan(const float* __restrict__ q, float* __restrict__ out) {
  const int lane = threadIdx.x & 31;
  const int wave = (blockIdx.x << 3) + (threadIdx.x >> 5);   // 0..4095
  const int n    = lane & 15;            // column within 16-col group
  const int half = lane >> 4;            // lane half (K / row split)
  const int m    = lane & 15;            // A-matrix row held by this lane

  const long col  = (long)wave * 16 + n; // flattened column = l*1024 + d
  const long l    = col >> 10;
  const long d    = col & 1023;
  const long outQ = l * (2 * kD) + d;    // within-turn offset, q half
  const long outA = outQ + kD;           // within-turn offset, avg half

  float carry = 0.0f;                    // running column sum of prior chunks

  for (int chunk = 0; chunk < kChunks; ++chunk) {
    const int t0 = chunk << 4;

    if (chunk + 1 < kChunks) {
      // hint the next chunk's rows into cache (global_prefetch_b8)
      __builtin_prefetch(q + (long)(t0 + 16 + 2 * half) * kC + col, 0, 0);
    }

    // ---- load 16x16 tile as four 4x16 B slices -------------------------
    // B (4x16 f32, 2 VGPRs): VGPR0 holds K = {0 | 2} across lane halves,
    // VGPR1 holds K = {1 | 3} (mirrors documented 16x4 A striping).
    v8f acc = {};
    float b0s[4], b1s[4];
#pragma unroll
    for (int kk = 0; kk < 4; ++kk) {
      const int r0 = (kk << 2) + (half << 1);          // turn row for b[0]
      const float b0 =
          __builtin_nontemporal_load(q + (long)(t0 + r0)     * kC + col);
      const float b1 =
          __builtin_nontemporal_load(q + (long)(t0 + r0 + 1) * kC + col);
      b0s[kk] = b0; b1s[kk] = b1;

      // A slice (16x4): strict lower triangle -> exclusive prefix
      v2f a = { (r0     < m) ? 1.0f : 0.0f,
                (r0 + 1 < m) ? 1.0f : 0.0f };
      v2f b = { b0, b1 };
      acc = __builtin_amdgcn_wmma_f32_16x16x4_f32(
          /*neg_a=*/false, a, /*neg_b=*/false, b,
          /*c_mod=*/(short)0, acc, /*reuse_a=*/false, /*reuse_b=*/false);
    }

    // ---- pass-through copy of q into out[:, :, 0:D] --------------------
#pragma unroll
    for (int kk = 0; kk < 4; ++kk) {
      const int r0 = (kk << 2) + (half << 1);
      __builtin_nontemporal_store(b0s[kk],
          out + (long)(t0 + r0)     * kOutRow + outQ);
      __builtin_nontemporal_store(b1s[kk],
          out + (long)(t0 + r0 + 1) * kOutRow + outQ);
    }

    // ---- avg: (carry + exclusive_prefix) * rcp(max(t,1)) ---------------
    // C/D layout: lanes 0-15 hold rows v, lanes 16-31 hold rows v+8.
    // At t==0 the exclusive prefix is exactly 0, so any finite reciprocal
    // gives the required avg[0]==0 -> no branch, no divide sequence.
#pragma unroll
    for (int v = 0; v < 8; ++v) {
      const int   t    = t0 + v + (half << 3);
      const int   te   = (t > 0) ? t : 1;              // v_max_i32
      const float excl = carry + acc[v];
      const float inv  = __builtin_amdgcn_rcpf((float)te);  // v_rcp_f32
      __builtin_nontemporal_store(excl * inv,
          out + (long)t * kOutRow + outA);
    }

    // ---- carry update: column total = excl[15] + x[15] -----------------
    // Row 15 lives in lanes 16-31 (acc[7]); x[15] was b1s[3] on those lanes.
    const float tail = acc[7] + b1s[3];
    carry += __shfl(tail, n + 16, 32);   // broadcast from the half==1 lane
  }
}

extern "C" void kernel_launch(void* const* d_in, const int* in_sizes, int n_in,
                              void* d_out, int out_size, void* d_ws, size_t ws_size,
                              hipStream_t stream) {
  const float* q = (const float*)d_in[0];
  float* out     = (float*)d_out;
  (void)in_sizes; (void)n_in; (void)out_size; (void)d_ws; (void)ws_size;

  dim3 grid(kWaves / 8);   // 512 blocks
  dim3 block(256);         // 8 wave32 per block
  hipLaunchKernelGGL(qflow_wmma_scan, grid, block, 0, stream, q, out);
}